// ReadHead_420906795794
// MI455X (gfx1250) — compile-verified
//
#include <hip/hip_runtime.h>
#include <hip/hip_bf16.h>
#include <math.h>

// ReadHead (DNC-style) for MI455X / gfx1250, wave32, fp32 WMMA path.
// Shapes: B=64, H=8, N=1024, M=128.

typedef __attribute__((ext_vector_type(2))) float v2f;
typedef __attribute__((ext_vector_type(8))) float v8f;

#define B_ 64
#define H_ 8
#define N_ 1024
#define M_ 128
#define EPS_ 1e-6f
#define MASK_MIN_ 0.0f

// Workspace layout (in floats):
//   W1 packed: [b][k/2][16][2]  -> B*64*16*2 = 131072
//   W2 packed: same             -> 131072
//   scores:    [b][h][n] h<8    -> B*8*N     = 524288
//   read_dist: [b][16][n] (rows 8..15 zero) -> B*16*N = 1048576
#define WS_W1 0
#define WS_W2 (B_ * 2048)
#define WS_SC (2 * B_ * 2048)
#define WS_RD (WS_SC + (size_t)B_ * H_ * N_)

// ---------------------------------------------------------------------------
// Kernel 1: per (b,h) build the two GEMM weight vectors.
//   w1[m] = mask[m] * khat[m]   (khat = normalized masked key)
//   w2[m] = mask[m]^2
// Stored K-pair packed so the WMMA B-fragment is a contiguous b64 per lane.
// Columns h = 8..15 are zeroed (padding for the 16-wide WMMA tile).
// ---------------------------------------------------------------------------
__global__ __launch_bounds__(128) void prep_kernel(
    const float* __restrict__ keys, const float* __restrict__ mask_in,
    float* __restrict__ ws)
{
    const int bh = blockIdx.x;          // 0 .. B*16-1
    const int b  = bh >> 4;
    const int h  = bh & 15;
    const int m  = threadIdx.x;         // 0..127 (= K index)
    const int p  = m >> 1;              // K pair
    const int j  = m & 1;

    float* w1 = ws + WS_W1;
    float* w2 = ws + WS_W2;
    const size_t widx = (((size_t)b * 64 + p) * 16 + h) * 2 + j;

    if (h >= H_) {                       // zero padding columns
        w1[widx] = 0.0f;
        w2[widx] = 0.0f;
        return;
    }

    float mk = mask_in[((size_t)b * H_ + h) * M_ + m];
    mk = mk * (1.0f - MASK_MIN_) + MASK_MIN_;
    const float km = keys[((size_t)b * H_ + h) * M_ + m] * mk;

    __shared__ float red[128];
    red[m] = km * km;
    __syncthreads();
    for (int s = 64; s > 0; s >>= 1) {
        if (m < s) red[m] += red[m + s];
        __syncthreads();
    }
    const float norm = sqrtf(red[0]);
    const float khat = km / (norm + EPS_);

    w1[widx] = mk * khat;
    w2[widx] = mk * mk;
}

// ---------------------------------------------------------------------------
// Kernel 2: content scores via two fused f32 WMMA GEMMs.
// One wave per 16-row N tile. K = 128 swept in 32 steps of 4:
//   accD += A   * W1   (masked-key dot product)
//   accN += A^2 * W2   (masked row squared norm)
// Then scores[n,h] = accD / (sqrt(accN)+eps) * beta[h].
//
// f32 16x16x4 operand layouts (ISA 7.12.2, wave32):
//   A: lane = {khalf,rowIdx}: v[j] = A[row = lane&15][k = 2*(lane>>4)+j]
//   B: v[j] = B[k = 2*(lane>>4)+j][col = lane&15]   (pair-packed in ws)
//   C/D: v[j] = D[row = j + 8*(lane>>4)][col = lane&15]
// ---------------------------------------------------------------------------
__global__ __launch_bounds__(128) void scores_kernel(
    const float* __restrict__ memory, const float* __restrict__ betas,
    float* __restrict__ ws)
{
    const int wave = threadIdx.x >> 5;
    const int lane = threadIdx.x & 31;
    const int idx  = lane & 15;          // A-row offset / B-col (=h)
    const int half = lane >> 4;          // k-pair select / D-row half
    const int b    = blockIdx.x >> 4;
    const int tile = ((blockIdx.x & 15) << 2) + wave;   // 0..63
    const int n0   = tile << 4;

    const float* memB = memory + (size_t)b * N_ * M_;
    const float* arow = memB + (size_t)(n0 + idx) * M_ + 2 * half;
    const float* w1   = ws + WS_W1 + (size_t)b * 2048 + (size_t)idx * 2;
    const float* w2   = ws + WS_W2 + (size_t)b * 2048 + (size_t)idx * 2;

    v8f accD = {};
    v8f accN = {};
    for (int kk = 0; kk < 32; ++kk) {
        const v2f a  = *(const v2f*)(arow + kk * 4);
        const v2f a2 = a * a;
        const int p  = kk * 2 + half;            // K pair index
        const v2f b1 = *(const v2f*)(w1 + (size_t)p * 32);
        const v2f b2 = *(const v2f*)(w2 + (size_t)p * 32);
        accD = __builtin_amdgcn_wmma_f32_16x16x4_f32(
            false, a,  false, b1, (short)0, accD, false, false);
        accN = __builtin_amdgcn_wmma_f32_16x16x4_f32(
            false, a2, false, b2, (short)0, accN, false, false);
    }

    if (idx < H_) {                      // only real heads
        const float beta = betas[(size_t)b * H_ + idx];
        float* sc = ws + WS_SC + ((size_t)b * H_ + idx) * N_ + n0 + 8 * half;
#pragma unroll
        for (int j = 0; j < 8; ++j) {
            sc[j] = accD[j] / (sqrtf(accN[j]) + EPS_) * beta;
        }
    }
}

// ---------------------------------------------------------------------------
// Kernel 3: softmax over N + gated 3-way mix -> read_dist, zero-padded to 16
// rows per batch so the final WMMA GEMM needs no predication.
// ---------------------------------------------------------------------------
__global__ __launch_bounds__(256) void mix_kernel(
    const float* __restrict__ fwd, const float* __restrict__ bwd,
    const float* __restrict__ gates, float* __restrict__ ws)
{
    const int bh = blockIdx.x;           // 0 .. B*16-1
    const int b  = bh >> 4;
    const int h  = bh & 15;
    const int t  = threadIdx.x;

    float* rd = ws + WS_RD + ((size_t)b * 16 + h) * N_;
    if (h >= H_) {                        // padding rows: zero
        for (int n = t; n < N_; n += 256) rd[n] = 0.0f;
        return;
    }

    const float* sc = ws + WS_SC + ((size_t)b * H_ + h) * N_;
    __shared__ float red[256];

    float v[4];
    float lmax = -INFINITY;
#pragma unroll
    for (int i = 0; i < 4; ++i) {
        v[i] = sc[t + i * 256];
        lmax = fmaxf(lmax, v[i]);
    }
    red[t] = lmax;
    __syncthreads();
    for (int s = 128; s > 0; s >>= 1) {
        if (t < s) red[t] = fmaxf(red[t], red[t + s]);
        __syncthreads();
    }
    const float mx = red[0];
    __syncthreads();

    float lsum = 0.0f;
#pragma unroll
    for (int i = 0; i < 4; ++i) {
        v[i] = expf(v[i] - mx);
        lsum += v[i];
    }
    red[t] = lsum;
    __syncthreads();
    for (int s = 128; s > 0; s >>= 1) {
        if (t < s) red[t] += red[t + s];
        __syncthreads();
    }
    const float inv = 1.0f / red[0];

    const size_t gbase = ((size_t)b * H_ + h) * 3;
    const float g0 = gates[gbase + 0];   // backward
    const float g1 = gates[gbase + 1];   // content
    const float g2 = gates[gbase + 2];   // forward
    const float* bw = bwd + ((size_t)b * H_ + h) * N_;
    const float* fw = fwd + ((size_t)b * H_ + h) * N_;
#pragma unroll
    for (int i = 0; i < 4; ++i) {
        const int n = t + i * 256;
        rd[n] = bw[n] * g0 + v[i] * inv * g1 + fw[n] * g2;
    }
}

// ---------------------------------------------------------------------------
// Kernel 4: read_data[b,h,m] = sum_n read_dist[b,h,n] * memory[b,n,m]
// via f32 WMMA: A = read_dist (16xN, rows 8..15 zero), B = memory (NxM),
// one wave per 16-wide M tile, 8 waves per batch, K=1024 in steps of 4.
// ---------------------------------------------------------------------------
__global__ __launch_bounds__(256) void readdata_kernel(
    const float* __restrict__ memory, const float* __restrict__ ws_in,
    float* __restrict__ out)
{
    const int wave = threadIdx.x >> 5;   // M tile 0..7
    const int lane = threadIdx.x & 31;
    const int idx  = lane & 15;          // A-row (=h padded) / B-col
    const int half = lane >> 4;          // k-pair select / D-row half
    const int b    = blockIdx.x;
    const int m0   = wave << 4;

    const float* memB = memory + (size_t)b * N_ * M_ + m0 + idx;
    const float* rd   = ws_in + WS_RD + ((size_t)b * 16 + idx) * N_ + 2 * half;

    v8f acc = {};
    for (int k0 = 0; k0 < N_; k0 += 4) {
        const v2f a = *(const v2f*)(rd + k0);
        v2f bf;
        bf[0] = memB[(size_t)(k0 + 2 * half + 0) * M_];
        bf[1] = memB[(size_t)(k0 + 2 * half + 1) * M_];
        acc = __builtin_amdgcn_wmma_f32_16x16x4_f32(
            false, a, false, bf, (short)0, acc, false, false);
    }

    if (half == 0) {                     // D rows 0..7 == heads 0..7
#pragma unroll
        for (int j = 0; j < 8; ++j) {
            out[((size_t)b * H_ + j) * M_ + m0 + idx] = acc[j];
        }
    }
}

// ---------------------------------------------------------------------------
extern "C" void kernel_launch(void* const* d_in, const int* in_sizes, int n_in,
                              void* d_out, int out_size, void* d_ws, size_t ws_size,
                              hipStream_t stream)
{
    const float* memory = (const float*)d_in[0];   // [B,N,M]
    const float* keys   = (const float*)d_in[1];   // [B,H,M]
    const float* betas  = (const float*)d_in[2];   // [B,H]
    const float* fwd    = (const float*)d_in[3];   // [B,H,N]
    const float* bwd    = (const float*)d_in[4];   // [B,H,N]
    const float* gates  = (const float*)d_in[5];   // [B,H,3]
    const float* maskp  = (const float*)d_in[6];   // [B,H,M]
    float* out = (float*)d_out;                    // [B,H,M]
    float* ws  = (float*)d_ws;                     // >= ~7.1 MB of floats

    prep_kernel    <<<B_ * 16, 128, 0, stream>>>(keys, maskp, ws);
    scores_kernel  <<<B_ * 16, 128, 0, stream>>>(memory, betas, ws);
    mix_kernel     <<<B_ * 16, 256, 0, stream>>>(fwd, bwd, gates, ws);
    readdata_kernel<<<B_,      256, 0, stream>>>(memory, ws, out);
}